// CamPoseNet_86131274154501
// MI455X (gfx1250) — compile-verified
//
#include <hip/hip_runtime.h>
#include <stdint.h>

// ---------------------------------------------------------------------------
// CamPoseNet sampler + quaternion transform for MI455X (gfx1250, wave32).
//
// Roofline: 44 B/row * 1M rows = 44 MB @ 23.3 TB/s => ~1.9 us. HBM-bound.
// The only matrix op, out[n] = E(q_n)^T X_n, is reformulated as
//   Out[16x4] = sum_g A_g[16x4] x B_g[4x16]     (g = 0..3)
// with A_g[m][k] = X_g(row m) * q_k(row m) and B_g a constant +-1 sign
// matrix, executed as 4 chained V_WMMA_F32_16X16X4_F32 per 16 rows (f32
// precision preserved; matrix-pipe time is free under the HBM bound).
// B is synthesized branch-free from packed immediates (no memory traffic,
// no exec-mask divergence between the WMMAs).  Full tiles take an
// unguarded store fast path (wave-uniform branch); 32-bit index math.
// ---------------------------------------------------------------------------

typedef __attribute__((ext_vector_type(2))) float v2f;
typedef __attribute__((ext_vector_type(8))) float v8f;

__device__ __forceinline__ uint32_t pcg(uint32_t x) {
  x = x * 747796405u + 2891336453u;
  uint32_t w = ((x >> ((x >> 28u) + 4u)) ^ x) * 277803737u;
  return (w >> 22u) ^ w;
}

// uniform in (0,1), never 0 (safe for log)
__device__ __forceinline__ float u01(uint32_t h) {
  return (float)(h >> 8) * (1.0f / 16777216.0f) + (0.5f / 16777216.0f);
}

// Decode 2-bit sign code at column n from packed constant: 01 -> +1, 10 -> -1.
__device__ __forceinline__ float bsign(uint32_t packed, uint32_t n) {
  const uint32_t bits = (packed >> (2u * n)) & 3u;
  return (float)(int)(bits & 1u) - (float)(int)(bits >> 1u);
}

__global__ __launch_bounds__(256)
void campose_bingham_kernel(const float* __restrict__ q,
                            const float* __restrict__ Z,
                            const int*   __restrict__ seedp,
                            float*       __restrict__ out,
                            int N)
{
  const uint32_t lane = threadIdx.x & 31u;
  const uint32_t waveInBlock = threadIdx.x >> 5;
  const uint32_t waveIdx = blockIdx.x * (blockDim.x >> 5) + waveInBlock;
  const uint32_t rowbase = waveIdx * 16u;     // 16 rows per wave (duplicated halves)
  if (rowbase >= (uint32_t)N) return;         // wave-uniform branch: EXEC stays full

  const uint32_t m = lane & 15u;              // row-in-tile; lanes L and L+16 share a row
  uint32_t row = rowbase + m;
  if (row > (uint32_t)N - 1u) row = (uint32_t)N - 1u;  // clamp (select, not branch)

  // ---- normalize quaternion (coalesced 16B load) ----
  const float4 qv = ((const float4*)q)[row];
  const float rinv = rsqrtf(qv.x*qv.x + qv.y*qv.y + qv.z*qv.z + qv.w*qv.w);
  float qn[4] = { qv.x*rinv, qv.y*rinv, qv.z*rinv, qv.w*rinv };

  // ---- lam = -Zexp, Zexp = [-1e-6, Zbatch] ----
  float lam[4];
  lam[0] = 1e-6f;
  lam[1] = -Z[row*3u + 0u];
  lam[2] = -Z[row*3u + 1u];
  lam[3] = -Z[row*3u + 2u];
  float sa[4], sig[4];
#pragma unroll
  for (int i = 0; i < 4; ++i) {
    sa[i]  = 1.0f + 2.0f * lam[i];     // sigacginv
    sig[i] = rsqrtf(sa[i]);            // sqrt(1/sigacginv)
  }

  // ---- ACG rejection sampler for Bingham(lam) ----
  const uint32_t seedv = (uint32_t)seedp[0];
  const uint32_t rbase = pcg(seedv + 0x9E3779B9u) ^ (row * 0x85EBCA6Bu);
  const float TWO_PI = 6.28318530717958647692f;
  const float CONST_TERM = 1.5f - 2.0f * 1.38629436111989062f; // 0.5*(QA-1) - (QA/2)*ln(QA)

  float X[4] = {0.0f, 0.0f, 0.0f, 0.0f};
  for (int it = 0; it < 256; ++it) {
    const uint32_t c0 = pcg(rbase + (uint32_t)it * 0x632BE5ABu);
    const float u1 = u01(pcg(c0 + 1u));
    const float u2 = u01(pcg(c0 + 2u));
    const float u3 = u01(pcg(c0 + 3u));
    const float u4 = u01(pcg(c0 + 4u));
    const float uu = u01(pcg(c0 + 5u));
    // Box-Muller -> 4 standard normals
    const float r1 = sqrtf(-2.0f * __logf(u1));
    const float r2 = sqrtf(-2.0f * __logf(u3));
    float s1, cc1, s2, cc2;
    __sincosf(TWO_PI * u2, &s1, &cc1);
    __sincosf(TWO_PI * u4, &s2, &cc2);
    const float yp0 = r1 * cc1 * sig[0];
    const float yp1 = r1 * s1  * sig[1];
    const float yp2 = r2 * cc2 * sig[2];
    const float yp3 = r2 * s2  * sig[3];
    const float inv = rsqrtf(yp0*yp0 + yp1*yp1 + yp2*yp2 + yp3*yp3);
    const float y0 = yp0*inv, y1 = yp1*inv, y2 = yp2*inv, y3 = yp3*inv;
    const float y20 = y0*y0, y21 = y1*y1, y22 = y2*y2, y23 = y3*y3;
    const float dlam = y20*lam[0] + y21*lam[1] + y22*lam[2] + y23*lam[3];
    const float dsa  = y20*sa[0]  + y21*sa[1]  + y22*sa[2]  + y23*sa[3];
    const float lratio = -dlam + 2.0f * __logf(dsa) + CONST_TERM;
    if (__logf(uu) < lratio) { X[0]=y0; X[1]=y1; X[2]=y2; X[3]=y3; break; }
  }
  // (control flow reconverges here; EXEC all-1s again for WMMA)

  // ---- out[16x4] = sum_g A_g[16x4] * B_g[4x16] via chained f32 WMMA ----
  // A 16x4 f32 layout: lanes 0-15 hold K=0 (vgpr0) / K=1 (vgpr1) of row M=lane;
  // lanes 16-31 hold K=2 / K=3 of row M=lane-16.  Lane-local since both
  // half-wave lanes computed the same row.
  //
  // B sign rows packed as immediates, 2 bits per output column j:
  //   code 01 = +1, code 10 = -1, 00 = 0.  Bc[g][k] encodes W[4g+k][0..3].
  // Derived from E rows [q0,-q1,-q2,q3],[q1,q0,q3,q2],[q2,-q3,q0,-q1],[q3,q2,-q1,-q0].
  const bool lowHalf = (lane < 16u);
  const int khalf = lowHalf ? 0 : 2;
  v8f cacc = {0.0f, 0.0f, 0.0f, 0.0f, 0.0f, 0.0f, 0.0f, 0.0f};

#if __has_builtin(__builtin_amdgcn_wmma_f32_16x16x4_f32)
  // Bc[g][k]:
  //  g=0: {0x01, 0x08, 0x20, 0x40}
  //  g=1: {0x04, 0x01, 0x40, 0x10}
  //  g=2: {0x10, 0x80, 0x01, 0x08}
  //  g=3: {0x80, 0x20, 0x04, 0x01}
  const uint32_t ck0_g[4] = { lowHalf ? 0x01u : 0x20u,
                              lowHalf ? 0x04u : 0x40u,
                              lowHalf ? 0x10u : 0x01u,
                              lowHalf ? 0x80u : 0x04u };
  const uint32_t ck1_g[4] = { lowHalf ? 0x08u : 0x40u,
                              lowHalf ? 0x01u : 0x10u,
                              lowHalf ? 0x80u : 0x08u,
                              lowHalf ? 0x20u : 0x01u };
#pragma unroll
  for (int g = 0; g < 4; ++g) {
    v2f a, b;
    a.x = X[g] * qn[khalf + 0];
    a.y = X[g] * qn[khalf + 1];
    b.x = bsign(ck0_g[g], m);   // zero for m >= 4 (codes only occupy bits [7:0])
    b.y = bsign(ck1_g[g], m);
    cacc = __builtin_amdgcn_wmma_f32_16x16x4_f32(
        /*neg_a=*/false, a, /*neg_b=*/false, b,
        /*c_mod=*/(short)0, cacc, /*reuse_a=*/false, /*reuse_b=*/false);
  }
  // D layout: vgpr v, lanes 0-15 -> (M=v, N=lane); lanes 16-31 -> (M=v+8, N=lane-16).
  // Valid output columns are N=0..3.  Full tiles (the common case; every tile
  // when N % 16 == 0) take the unguarded path via one wave-uniform branch.
  if (m < 4u) {
    const uint32_t mb = lowHalf ? 0u : 8u;
    const uint32_t base = (rowbase + mb) * 4u + m;
    if (rowbase + 16u <= (uint32_t)N) {
#pragma unroll
      for (int v = 0; v < 8; ++v) out[base + 4u * (uint32_t)v] = cacc[v];
    } else {
#pragma unroll
      for (int v = 0; v < 8; ++v) {
        if (rowbase + mb + (uint32_t)v < (uint32_t)N) out[base + 4u * (uint32_t)v] = cacc[v];
      }
    }
  }
#else
  // Fallback (should not be taken on gfx1250): direct 16-FMA transform.
  const float o0 =  qn[0]*X[0] + qn[1]*X[1] + qn[2]*X[2] + qn[3]*X[3];
  const float o1 = -qn[1]*X[0] + qn[0]*X[1] - qn[3]*X[2] + qn[2]*X[3];
  const float o2 = -qn[2]*X[0] + qn[3]*X[1] + qn[0]*X[2] - qn[1]*X[3];
  const float o3 =  qn[3]*X[0] + qn[2]*X[1] - qn[1]*X[2] - qn[0]*X[3];
  if (lowHalf && (rowbase + m) < (uint32_t)N) {
    ((float4*)out)[row] = make_float4(o0, o1, o2, o3);
  }
#endif
}

extern "C" void kernel_launch(void* const* d_in, const int* in_sizes, int n_in,
                              void* d_out, int out_size, void* d_ws, size_t ws_size,
                              hipStream_t stream) {
  const float* q    = (const float*)d_in[0];   // [N,4] f32
  const float* Z    = (const float*)d_in[1];   // [N,3] f32
  const int*   seed = (const int*)d_in[2];     // scalar
  float* out = (float*)d_out;                  // [N,4] f32

  const int N = in_sizes[0] / 4;
  const int wavesNeeded  = (N + 15) / 16;      // 16 rows per wave
  const int wavesPerBlk  = 256 / 32;           // 8 waves per 256-thread block
  const int blocks = (wavesNeeded + wavesPerBlk - 1) / wavesPerBlk;

  hipLaunchKernelGGL(campose_bingham_kernel, dim3(blocks), dim3(256), 0, stream,
                     q, Z, seed, out, N);
}